// TopK_2662879723713
// MI455X (gfx1250) — compile-verified
//
#include <hip/hip_runtime.h>
#include <hip/hip_bf16.h>

#define COLS 16384
#define BLOCK 1024

typedef unsigned int u32x4 __attribute__((ext_vector_type(4)));
typedef int          i32x8 __attribute__((ext_vector_type(8)));
typedef int          i32x4 __attribute__((ext_vector_type(4)));

// Monotone map: larger float <-> larger unsigned key (no NaNs in input).
__device__ __forceinline__ unsigned f2key(float f) {
    unsigned u = __float_as_uint(f);
    return (u & 0x80000000u) ? ~u : (u | 0x80000000u);
}
__device__ __forceinline__ float key2f(unsigned k) {
    unsigned u = (k & 0x80000000u) ? (k & 0x7FFFFFFFu) : ~k;
    return __uint_as_float(u);
}

__global__ __launch_bounds__(BLOCK)
void topk_thresh_kernel(const float* __restrict__ x,
                        const int* __restrict__ kptr,
                        float* __restrict__ out) {
    __shared__ float    s_row[COLS];       // 64 KB: whole row staged in LDS
    __shared__ unsigned s_hist[256];
    __shared__ unsigned s_prefix;
    __shared__ int      s_remaining;

    const int row = blockIdx.x;
    const int tid = threadIdx.x;
    const float* __restrict__ xrow = x   + (size_t)row * COLS;
    float*       __restrict__ orow = out + (size_t)row * COLS;

    // ---- Stage the row into LDS via the Tensor Data Mover (one 64KB DMA) ----
#if defined(__gfx1250__) && __has_builtin(__builtin_amdgcn_tensor_load_to_lds)
    if (tid == 0) {
        const unsigned long long ga = (unsigned long long)(const void*)xrow;
        const unsigned lds_off = (unsigned)(unsigned long long)(void*)&s_row[0];
        // D# group 0 (ISA 8.3): count=1 valid, lds_addr, global_addr[56:0], type=2
        u32x4 g0;
        g0.x = 1u;                                   // count=1, is_restore=0, gather off
        g0.y = lds_off;                              // lds_addr (bytes)
        g0.z = (unsigned)(ga & 0xFFFFFFFFu);         // global_addr[31:0]
        g0.w = (unsigned)((ga >> 32) & 0x01FFFFFFu)  // global_addr[56:32]
             | (2u << 30);                           // type = 2 ("image")
        // D# group 1 (ISA 8.4): 1-D tile, data_size=4B, dim0 = tile_dim0 = COLS
        i32x8 g1;
        g1[0] = (int)(2u << 16);                         // wg_mask=0, data_size=2 (4B)
        g1[1] = (int)(((unsigned)COLS & 0xFFFFu) << 16); // tensor_dim0[15:0]
        g1[2] = (int)((((unsigned)COLS >> 16) & 0xFFFFu) // tensor_dim0[31:16]
                      | (1u << 16));                     // tensor_dim1 = 1
        g1[3] = (int)(((unsigned)COLS & 0xFFFFu) << 16); // tile_dim0 = COLS
        g1[4] = 1;                                       // tile_dim1=1, tile_dim2=0
        g1[5] = COLS;                                    // tensor_dim0_stride[31:0]
        g1[6] = (int)(((unsigned)COLS & 0xFFFFu) << 16); // dim0_stride hi=0 | dim1_stride[15:0]
        g1[7] = (int)(((unsigned)COLS >> 16) & 0xFFFFu); // dim1_stride[47:16]
        i32x4 gz4 = {0, 0, 0, 0};                        // groups 2/3 unused (<=2D)
        i32x8 gz8 = {0, 0, 0, 0, 0, 0, 0, 0};
        __builtin_amdgcn_tensor_load_to_lds(g0, g1, gz4, gz4, gz8, 0);
        __builtin_amdgcn_s_wait_tensorcnt(0);            // TENSORcnt == 0
    }
    __syncthreads();
#else
    for (int i = tid * 4; i < COLS; i += BLOCK * 4) {
        const float4 v = *(const float4*)(xrow + i);
        *(float4*)(&s_row[i]) = v;
    }
    __syncthreads();
#endif

    // ---- Exact k-th largest via 4-pass radix select on 8-bit digits ----
    if (tid == 0) {
        s_prefix = 0u;
        s_remaining = *kptr;   // k (device scalar)
    }
    __syncthreads();

    for (int pass = 0; pass < 4; ++pass) {
        const int shift = 24 - 8 * pass;
        const unsigned mask = (pass == 0) ? 0u : (0xFFFFFFFFu << (32 - 8 * pass));

        for (int i = tid; i < 256; i += BLOCK) s_hist[i] = 0u;
        __syncthreads();

        const unsigned cur_prefix = s_prefix;
        for (int i = tid; i < COLS; i += BLOCK) {          // bank-conflict-free b32
            const unsigned key = f2key(s_row[i]);
            if ((key & mask) == cur_prefix)
                atomicAdd(&s_hist[(key >> shift) & 0xFFu], 1u);
        }
        __syncthreads();

        if (tid == 0) {
            const int rem = s_remaining;
            unsigned cum = 0;
            int d = 255;
            for (; d > 0; --d) {                            // descending cumulative scan
                const unsigned c = s_hist[d];
                if (cum + c >= (unsigned)rem) break;
                cum += c;
            }
            s_remaining = rem - (int)cum;                   // rank within chosen bin
            s_prefix    = cur_prefix | ((unsigned)d << shift);
        }
        __syncthreads();
    }

    const float thresh = key2f(s_prefix);

    // ---- Threshold and stream out (128-bit stores, row read from LDS) ----
    for (int i = tid * 4; i < COLS; i += BLOCK * 4) {
        const float4 v = *(const float4*)(&s_row[i]);
        float4 o;
        o.x = (v.x >= thresh) ? v.x : 0.0f;
        o.y = (v.y >= thresh) ? v.y : 0.0f;
        o.z = (v.z >= thresh) ? v.z : 0.0f;
        o.w = (v.w >= thresh) ? v.w : 0.0f;
        *(float4*)(orow + i) = o;
    }
}

extern "C" void kernel_launch(void* const* d_in, const int* in_sizes, int n_in,
                              void* d_out, int out_size, void* d_ws, size_t ws_size,
                              hipStream_t stream) {
    const float* x    = (const float*)d_in[0];
    const int*   kptr = (const int*)d_in[1];
    float*       out  = (float*)d_out;

    const int rows = in_sizes[0] / COLS;   // 8192 for the reference shape
    topk_thresh_kernel<<<dim3(rows), dim3(BLOCK), 0, stream>>>(x, kptr, out);
}